// TopkSparseAttention_87952340287566
// MI455X (gfx1250) — compile-verified
//
#include <hip/hip_runtime.h>
#include <stdint.h>

#define TOPK    17
#define N_COLS  8192
#define BLOCK   256
#define PER_T   8   // float4 loads per thread -> 32 elements/thread * 256 threads = 8192

typedef unsigned int u32x4 __attribute__((ext_vector_type(4)));
typedef int          i32x4 __attribute__((ext_vector_type(4)));
typedef int          i32x8 __attribute__((ext_vector_type(8)));

// Monotone (order-preserving) float32 -> uint32 key
__device__ __forceinline__ unsigned f2key(float f) {
    unsigned u = __float_as_uint(f);
    return (u & 0x80000000u) ? ~u : (u | 0x80000000u);
}

__global__ __launch_bounds__(BLOCK)
void topk_sparse_rownorm(const float* __restrict__ attn, float* __restrict__ out) {
    const int tid  = threadIdx.x;
    const int row  = blockIdx.x;
    const int lane = tid & 31;
    const int wave = tid >> 5;

    const float* rowIn  = attn + (size_t)row * N_COLS;
    float*       rowOut = out  + (size_t)row * N_COLS;

    __shared__ unsigned s_hist[8 * 256];   // one 256-bin histogram per wave
    __shared__ unsigned s_selwant[2];      // [0]=selected bin, [1]=residual rank
    __shared__ float    s_red[8];
    __shared__ float    s_denom;

    float4 v[PER_T];

#if __has_builtin(__builtin_amdgcn_tensor_load_to_lds) && __has_builtin(__builtin_amdgcn_s_wait_tensorcnt)
    // ---- TDM: single DMA moves the whole 32KB row HBM -> LDS ----
    __shared__ __align__(16) float s_row[N_COLS];
    if (tid < 32) {  // wave 0 only (uniform branch); TDM ignores EXEC
        unsigned ldsoff = (unsigned)(uintptr_t)(&s_row[0]);       // low 32 bits of flat LDS addr = LDS offset
        unsigned long long ga = (unsigned long long)(uintptr_t)rowIn;
        u32x4 g0;
        g0[0] = 1u;                                               // count=1, user descriptor
        g0[1] = ldsoff;                                           // lds_addr [63:32]
        g0[2] = (unsigned)ga;                                     // global_addr[31:0]
        g0[3] = ((unsigned)((ga >> 32) & 0x01FFFFFFull))          // global_addr[56:32]
              | 0x80000000u;                                      // type=2 ("image") at [127:126]
        i32x8 g1;
        g1[0] = 0x00020000;                                       // data_size=2 (4B); no multicast/pad/iterate
        g1[1] = (int)(((unsigned)N_COLS & 0xFFFFu) << 16);        // tensor_dim0[15:0] at bits [63:48]
        g1[2] = (int)(1u << 16);                                  // tensor_dim0[31:16]=0, tensor_dim1=1
        g1[3] = (int)((unsigned)N_COLS << 16);                    // tensor_dim1 hi=0, tile_dim0=8192
        g1[4] = 1;                                                // tile_dim1=1, tile_dim2=0
        g1[5] = N_COLS;                                           // tensor_dim0_stride[31:0]
        g1[6] = (int)(((unsigned)N_COLS & 0xFFFFu) << 16);        // dim0_stride hi=0, dim1_stride lo
        g1[7] = 0;
        i32x4 g2 = {0, 0, 0, 0};                                  // <=2D: groups 2/3 zero
        i32x4 g3 = {0, 0, 0, 0};
        i32x8 g4 = {0, 0, 0, 0, 0, 0, 0, 0};                      // reserved/extra operand bundle (6-arg form)
        __builtin_amdgcn_tensor_load_to_lds(g0, g1, g2, g3, g4, 0);
        __builtin_amdgcn_s_wait_tensorcnt(0);
    }
    __syncthreads();
    const float4* srcV4 = (const float4*)s_row;                   // ds_load_b128 into registers once
#else
    const float4* srcV4 = (const float4*)rowIn;                   // fallback: direct coalesced global loads
#endif

    #pragma unroll
    for (int i = 0; i < PER_T; ++i)
        v[i] = srcV4[(i << 8) + tid];

    // ---- 4-pass MSB-first radix select for the 17th-largest ----
    unsigned prefix = 0;
    unsigned want   = TOPK;

    #pragma unroll
    for (int pass = 0; pass < 4; ++pass) {
        const int shift = 24 - 8 * pass;

        for (int i = tid; i < 8 * 256; i += BLOCK) s_hist[i] = 0;
        __syncthreads();

        unsigned* myh = s_hist + (wave << 8);
        #pragma unroll
        for (int i = 0; i < PER_T; ++i) {
            const float fv[4] = {v[i].x, v[i].y, v[i].z, v[i].w};
            #pragma unroll
            for (int c = 0; c < 4; ++c) {
                unsigned u  = f2key(fv[c]);
                unsigned hi = (pass == 0) ? 0u : (u >> (unsigned)(shift + 8));
                unsigned pf = (pass == 0) ? 0u : prefix;
                if (hi == pf) atomicAdd(&myh[(u >> shift) & 255u], 1u);  // ds_add_u32
            }
        }
        __syncthreads();

        // combine the 8 per-wave copies into copy 0 (thread t owns bin t; no cross-thread hazard)
        unsigned csum = 0;
        #pragma unroll
        for (int w = 0; w < 8; ++w) csum += s_hist[(w << 8) + tid];
        s_hist[tid] = csum;
        __syncthreads();

        // wave32 parallel scan over 256 bins, high bins first: lane L owns bins [255-8L .. 248-8L]
        if (tid < 32) {
            const int base = 248 - (lane << 3);
            unsigned g[8], t = 0;
            #pragma unroll
            for (int j = 0; j < 8; ++j) { g[j] = s_hist[base + j]; t += g[j]; }
            unsigned pre = t;
            #pragma unroll
            for (int d = 1; d < 32; d <<= 1) {
                unsigned o = __shfl_up(pre, (unsigned)d, 32);
                if (lane >= d) pre += o;
            }
            const unsigned before = pre - t;      // count in bins above this lane's group
            if (before < want && want <= pre) {   // exactly one lane hits
                unsigned cum = before;
                #pragma unroll
                for (int j = 7; j >= 0; --j) {
                    cum += g[j];
                    if (cum >= want) {
                        s_selwant[0] = (unsigned)(base + j);
                        s_selwant[1] = want - (cum - g[j]);
                        break;
                    }
                }
            }
        }
        __syncthreads();
        prefix = (prefix << 8) | s_selwant[0];
        want   = s_selwant[1];
        __syncthreads();
    }

    // prefix is now the sortable key of the 17th-largest value
    const unsigned kb = (prefix & 0x80000000u) ? (prefix & 0x7FFFFFFFu) : ~prefix;
    const float eps    = 1.1920928955078125e-07f;   // FLT_EPSILON = 2^-23
    const float thresh = __uint_as_float(kb) + eps;

    // ---- row sum of relu(x - thresh): wave32 shuffle reduce + 8-wave LDS reduce ----
    float lsum = 0.0f;
    #pragma unroll
    for (int i = 0; i < PER_T; ++i) {
        lsum += fmaxf(v[i].x - thresh, 0.0f);
        lsum += fmaxf(v[i].y - thresh, 0.0f);
        lsum += fmaxf(v[i].z - thresh, 0.0f);
        lsum += fmaxf(v[i].w - thresh, 0.0f);
    }
    #pragma unroll
    for (int off = 16; off > 0; off >>= 1)
        lsum += __shfl_down(lsum, (unsigned)off, 32);
    if (lane == 0) s_red[wave] = lsum;
    __syncthreads();
    if (tid == 0) {
        float d = eps;
        #pragma unroll
        for (int w = 0; w < 8; ++w) d += s_red[w];
        s_denom = d;
    }
    __syncthreads();
    const float inv = 1.0f / s_denom;

    // ---- normalize + coalesced b128 stores ----
    #pragma unroll
    for (int i = 0; i < PER_T; ++i) {
        float4 o;
        o.x = fmaxf(v[i].x - thresh, 0.0f) * inv;
        o.y = fmaxf(v[i].y - thresh, 0.0f) * inv;
        o.z = fmaxf(v[i].z - thresh, 0.0f) * inv;
        o.w = fmaxf(v[i].w - thresh, 0.0f) * inv;
        ((float4*)rowOut)[(i << 8) + tid] = o;
    }
}

extern "C" void kernel_launch(void* const* d_in, const int* in_sizes, int n_in,
                              void* d_out, int out_size, void* d_ws, size_t ws_size,
                              hipStream_t stream) {
    (void)n_in; (void)out_size; (void)d_ws; (void)ws_size;
    const float* attn = (const float*)d_in[0];
    float* out = (float*)d_out;
    const int rows = in_sizes[0] / N_COLS;   // 8192 x 8192
    topk_sparse_rownorm<<<rows, BLOCK, 0, stream>>>(attn, out);
}